// L_color_73358041416354
// MI455X (gfx1250) — compile-verified
//
#include <hip/hip_runtime.h>
#include <hip/hip_bf16.h>

typedef __attribute__((ext_vector_type(2))) float v2f;
typedef __attribute__((ext_vector_type(8))) float v8f;

#define NBLK 6144          // 32 * 3 * 64 blocks of 128x128
#define W 1024
#define BLKROWS 128

// ---------------------------------------------------------------------------
// Kernel 1: per-block (128x128) sum/min/max. One workgroup per block.
// Sum is accumulated on the matrix pipe via V_WMMA_F32_16X16X4_F32 with
// A = ones(16x4): D[m,n] += sum_k B[k,n]  -> column partial sums in C.
// Since the block-sum is permutation invariant, the raw float4 register data
// can be fed directly as the B operand regardless of logical (k,n) placement.
// ---------------------------------------------------------------------------
__global__ __launch_bounds__(256)
void block_stats_kernel(const float* __restrict__ x, float* __restrict__ ws)
{
    const int bid  = blockIdx.x;          // 0..6143 : (b*3+c)*64 + i*8 + j
    const int bc   = bid >> 6;            // b*3+c
    const int blk  = bid & 63;
    const int bi   = blk >> 3;
    const int bj   = blk & 7;

    const float* __restrict__ base =
        x + (size_t)bc * (W * W) + (size_t)bi * (BLKROWS * W) + (size_t)bj * BLKROWS;

    const int lane = threadIdx.x & 31;    // column-group 0..31 (float4 index)
    const int wv   = threadIdx.x >> 5;    // wave 0..7 -> starting row

    v2f ones; ones[0] = 1.0f; ones[1] = 1.0f;
    v8f acc = {};                         // WMMA C accumulator (column sums)
    float vmin =  __builtin_inff();
    float vmax = -__builtin_inff();

    #pragma unroll 4
    for (int s = 0; s < 16; ++s) {
        const int row = (s << 3) + wv;    // rows wv, wv+8, ..., wv+120
        // 16B-aligned, 512B contiguous per wave -> global_load_b128
        float4 v = ((const float4*)(base + (size_t)row * W))[lane];

        vmin = fminf(vmin, fminf(fminf(v.x, v.y), fminf(v.z, v.w)));
        vmax = fmaxf(vmax, fmaxf(fmaxf(v.x, v.y), fmaxf(v.z, v.w)));

        v2f b0; b0[0] = v.x; b0[1] = v.y;
        v2f b1; b1[0] = v.z; b1[1] = v.w;
        // D = ones(16x4) x B(4x16) + C  : accumulates 64 elements per call
        acc = __builtin_amdgcn_wmma_f32_16x16x4_f32(false, ones, false, b0,
                                                    (short)0, acc, false, false);
        acc = __builtin_amdgcn_wmma_f32_16x16x4_f32(false, ones, false, b1,
                                                    (short)0, acc, false, false);
    }

    // C[m,n] = colsum(n) replicated over m. Lane n and lane n+16 both hold
    // colsum(n) in VGPR0 of C -> cross-lane sum of acc[0] == 2 * block_sum.
    float s = acc[0];
    #pragma unroll
    for (int m = 16; m >= 1; m >>= 1) {
        s    += __shfl_xor(s, m, 32);
        vmin  = fminf(vmin, __shfl_xor(vmin, m, 32));
        vmax  = fmaxf(vmax, __shfl_xor(vmax, m, 32));
    }
    s *= 0.5f;

    __shared__ float ss[8], smn[8], smx[8];
    if (lane == 0) { ss[wv] = s; smn[wv] = vmin; smx[wv] = vmax; }
    __syncthreads();

    if (threadIdx.x == 0) {
        float S = 0.0f, mn = __builtin_inff(), mx = -__builtin_inff();
        #pragma unroll
        for (int k = 0; k < 8; ++k) {
            S += ss[k];
            mn = fminf(mn, smn[k]);
            mx = fmaxf(mx, smx[k]);
        }
        ws[bid]            = S * (1.0f / 16384.0f);  // mean
        ws[bid + NBLK]     = mn;
        ws[bid + 2 * NBLK] = mx;
    }
}

// ---------------------------------------------------------------------------
// Kernel 2: combine 2048 (b, block) positions * 3 channel pairs -> scalar
// ---------------------------------------------------------------------------
__global__ __launch_bounds__(256)
void finalize_kernel(const float* __restrict__ ws, float* __restrict__ out)
{
    const float* __restrict__ mean = ws;
    const float* __restrict__ mn   = ws + NBLK;
    const float* __restrict__ mx   = ws + 2 * NBLK;

    float acc = 0.0f;
    for (int idx = threadIdx.x; idx < 2048; idx += 256) {
        const int b   = idx >> 6;
        const int blk = idx & 63;
        const int i0  = (b * 3 + 0) * 64 + blk;
        const int i1  = i0 + 64;
        const int i2  = i0 + 128;

        const float m0 = mean[i0], m1 = mean[i1], m2 = mean[i2];
        const float n0 = mn[i0],   n1 = mn[i1],   n2 = mn[i2];
        const float x0 = mx[i0],   x1 = mx[i1],   x2 = mx[i2];

        const float d01 = ((n0 > x1) || (x0 < n1)) ? 0.0f : (m0 - m1) * (m0 - m1);
        const float d12 = ((n1 > x2) || (x1 < n2)) ? 0.0f : (m1 - m2) * (m1 - m2);
        const float d02 = ((n0 > x2) || (x0 < n2)) ? 0.0f : (m0 - m2) * (m0 - m2);
        acc += d01 + d12 + d02;
    }

    #pragma unroll
    for (int m = 16; m >= 1; m >>= 1) acc += __shfl_xor(acc, m, 32);

    __shared__ float part[8];
    const int lane = threadIdx.x & 31;
    const int wv   = threadIdx.x >> 5;
    if (lane == 0) part[wv] = acc;
    __syncthreads();

    if (threadIdx.x == 0) {
        float t = 0.0f;
        #pragma unroll
        for (int k = 0; k < 8; ++k) t += part[k];
        out[0] = t * (1.0f / 64.0f);
    }
}

extern "C" void kernel_launch(void* const* d_in, const int* in_sizes, int n_in,
                              void* d_out, int out_size, void* d_ws, size_t ws_size,
                              hipStream_t stream)
{
    const float* x  = (const float*)d_in[0];
    float* out      = (float*)d_out;
    float* ws       = (float*)d_ws;   // 3 * 6144 floats = 72 KB

    block_stats_kernel<<<dim3(NBLK), dim3(256), 0, stream>>>(x, ws);
    finalize_kernel<<<dim3(1), dim3(256), 0, stream>>>(ws, out);
}